// GAT_V2_24919400251447
// MI455X (gfx1250) — compile-verified
//
#include <hip/hip_runtime.h>
#include <stddef.h>

// ---------------------------------------------------------------------------
// GATv2 (2 layers) for MI455X / gfx1250, wave32.
//   Layer1: F_IN=23 -> H1=8 x C1=120 (D1=960), layer2: 960 -> 1.
//   Dense transforms use V_WMMA_F32_16X16X4_F32 (fp32 WMMA, K padded 23->24).
//   Edge phase is L2-resident gather/scatter (XL+XR = 153MB < 192MB L2);
//   gathers vectorized to b128 to maximize issue efficiency.
// ---------------------------------------------------------------------------

#define FIN   23
#define KPAD  24
#define D1    960
#define H1    8
#define C1    120
#define Q1    240          // D1 / 4 (float4s per row)
#define QH    30           // C1 / 4 (float4s per head)
#define NEG_SLOPE 0.2f

typedef __attribute__((ext_vector_type(2))) float v2f;
typedef __attribute__((ext_vector_type(8))) float v8f;

// order-preserving float <-> int map for atomicMax-based segment max
__device__ __forceinline__ int fmap_ord(float f) {
    int i = __float_as_int(f);
    return (i >= 0) ? i : (i ^ 0x7fffffff);
}
__device__ __forceinline__ float funmap_ord(int i) {
    return __int_as_float((i >= 0) ? i : (i ^ 0x7fffffff));
}

__device__ __forceinline__ float wave_sum(float v) {
    #pragma unroll
    for (int off = 16; off > 0; off >>= 1) v += __shfl_xor(v, off);
    return v;
}

__device__ __forceinline__ float lrelu(float v) {
    return (v > 0.0f) ? v : NEG_SLOPE * v;
}

__device__ __forceinline__ void edge_ends(const int* __restrict__ ei, int e, int E,
                                          int& s, int& d) {
    if (e < E) { s = ei[e]; d = ei[E + e]; }
    else       { s = e - E; d = e - E; }      // self loops appended
}

// ---------------------------------------------------------------------------
// init: zero accumulators, set segment-max buffers to -inf (mapped INT_MIN)
// ---------------------------------------------------------------------------
__global__ void gat_init(float4* __restrict__ out1, float* __restrict__ den1,
                         int* __restrict__ m1, float* __restrict__ den2,
                         int* __restrict__ m2, float* __restrict__ acc2, int N) {
    long long i = (long long)blockIdx.x * blockDim.x + threadIdx.x;
    long long tot4 = (long long)N * Q1;
    if (i < tot4)             out1[i] = make_float4(0.f, 0.f, 0.f, 0.f);
    if (i < (long long)N*H1) { den1[i] = 0.0f; m1[i] = (int)0x80000000; }
    if (i < N)               { den2[i] = 0.0f; m2[i] = (int)0x80000000; acc2[i] = 0.0f; }
}

// ---------------------------------------------------------------------------
// Layer-1 transforms: XL = x@W1l + b1l, XR = x@W1r + b1r via WMMA f32 16x16x4.
// Block = 128 threads (4 waves). Each wave owns a 16x16 output tile.
// blockIdx.y in [0,30): y<15 -> XL (W1l), else XR (W1r); 15*64 = 960 cols.
// A-tile (16 x 24, zero padded) staged in LDS, shared by the 4 waves.
// ---------------------------------------------------------------------------
__global__ __launch_bounds__(128) void gat_gemm_xlr(
    const float* __restrict__ x,
    const float* __restrict__ W1l, const float* __restrict__ b1l,
    const float* __restrict__ W1r, const float* __restrict__ b1r,
    float* __restrict__ XL, float* __restrict__ XR, int N)
{
    __shared__ float As[16][KPAD];
    const int rowbase = blockIdx.x * 16;
    const int t = threadIdx.x;

    for (int i = t; i < 16 * KPAD; i += 128) {
        int r = i / KPAD, k = i % KPAD;
        int gr = rowbase + r;
        As[r][k] = (k < FIN && gr < N) ? x[(size_t)gr * FIN + k] : 0.0f;
    }
    __syncthreads();

    const int wave = t >> 5;
    const int lane = t & 31;
    const int ytile = blockIdx.y;
    const bool right = (ytile >= 15);
    const float* __restrict__ W    = right ? W1r : W1l;
    const float* __restrict__ bias = right ? b1r : b1l;
    float* __restrict__ Out        = right ? XR : XL;

    const int col   = (ytile % 15) * 64 + wave * 16 + (lane & 15);
    const int khalf = (lane >> 4) * 2;          // 0 or 2
    const int m     = lane & 15;

    v8f c;
    const float bv = bias[col];
    #pragma unroll
    for (int i = 0; i < 8; ++i) c[i] = bv;

    #pragma unroll
    for (int kb = 0; kb < KPAD; kb += 4) {
        const int k0 = kb + khalf;
        v2f a, b;
        a.x = As[m][k0];
        a.y = As[m][k0 + 1];
        b.x = (k0     < FIN) ? W[(size_t)k0       * D1 + col] : 0.0f;
        b.y = (k0 + 1 < FIN) ? W[(size_t)(k0 + 1) * D1 + col] : 0.0f;
        c = __builtin_amdgcn_wmma_f32_16x16x4_f32(
                /*neg_a=*/false, a, /*neg_b=*/false, b,
                /*c_mod=*/(short)0, c, /*reuse_a=*/false, /*reuse_b=*/false);
    }

    const int rowoff = (lane >> 4) * 8;
    float* __restrict__ op = Out + (size_t)(rowbase + rowoff) * D1 + col;
    if (rowbase + 16 <= N) {            // full tile: straight store clause
        #pragma unroll
        for (int r = 0; r < 8; ++r)
            op[(size_t)r * D1] = c[r];
    } else {                            // ragged tail tile
        #pragma unroll
        for (int r = 0; r < 8; ++r)
            if (rowbase + rowoff + r < N) op[(size_t)r * D1] = c[r];
    }
}

// ---------------------------------------------------------------------------
// Layer-1 attention logits: one wave per edge.
// Per head: lanes 0..29 each take one float4 (4 channels) -> b128 gathers.
// e1[e][h] = sum_c att1[h][c] * leakyrelu(XL[s][h,c] + XR[d][h,c]); seg-max m1.
// ---------------------------------------------------------------------------
__global__ __launch_bounds__(256) void gat_edge_logits1(
    const float4* __restrict__ XL4, const float4* __restrict__ XR4,
    const float4* __restrict__ att4, const int* __restrict__ ei,
    float* __restrict__ e1, int* __restrict__ m1, int E, int ET)
{
    const int e = blockIdx.x * 8 + (threadIdx.x >> 5);
    if (e >= ET) return;
    const int lane = threadIdx.x & 31;
    int s, d; edge_ends(ei, e, E, s, d);
    const float4* __restrict__ xs = XL4 + (size_t)s * Q1;
    const float4* __restrict__ xd = XR4 + (size_t)d * Q1;
    #pragma unroll
    for (int h = 0; h < H1; ++h) {
        float acc = 0.0f;
        if (lane < QH) {
            const int i4 = h * QH + lane;
            float4 a = xs[i4], b = xd[i4], w = att4[i4];
            acc  = w.x * lrelu(a.x + b.x);
            acc += w.y * lrelu(a.y + b.y);
            acc += w.z * lrelu(a.z + b.z);
            acc += w.w * lrelu(a.w + b.w);
        }
        acc = wave_sum(acc);
        if (lane == 0) {
            e1[(size_t)e * H1 + h] = acc;
            atomicMax(&m1[d * H1 + h], fmap_ord(acc));
        }
    }
}

// exp(e - max) + segment denom; one thread per (edge, head)
__global__ void gat_edge_exp1(float* __restrict__ e1, const int* __restrict__ m1,
                              float* __restrict__ den1, const int* __restrict__ ei,
                              int E, int ET)
{
    long long t = (long long)blockIdx.x * blockDim.x + threadIdx.x;
    if (t >= (long long)ET * H1) return;
    int e = (int)(t >> 3), h = (int)(t & 7);
    int d = (e < E) ? ei[E + e] : e - E;
    float ex = __expf(e1[t] - funmap_ord(m1[d * H1 + h]));
    e1[t] = ex;
    __hip_atomic_fetch_add(&den1[d * H1 + h], ex,
                           __ATOMIC_RELAXED, __HIP_MEMORY_SCOPE_AGENT);
}

// out1[d] += alpha[e][h] * XL[s]; one wave per edge; b128 gathers, b32 atomics
__global__ __launch_bounds__(256) void gat_edge_scatter1(
    const float4* __restrict__ XL4, const float* __restrict__ e1,
    const float* __restrict__ den1, const int* __restrict__ ei,
    float* __restrict__ out1, int E, int ET)
{
    const int e = blockIdx.x * 8 + (threadIdx.x >> 5);
    if (e >= ET) return;
    const int lane = threadIdx.x & 31;
    int s, d; edge_ends(ei, e, E, s, d);
    float al[H1];
    #pragma unroll
    for (int h = 0; h < H1; ++h)
        al[h] = e1[(size_t)e * H1 + h] / (den1[d * H1 + h] + 1e-16f);
    const float4* __restrict__ xs = XL4 + (size_t)s * Q1;
    float* __restrict__ od = out1 + (size_t)d * D1;
    for (int i = lane; i < Q1; i += 32) {
        float4 v = xs[i];
        float  a = al[i / QH];
        float* p = od + 4 * i;
        __hip_atomic_fetch_add(p + 0, a * v.x, __ATOMIC_RELAXED, __HIP_MEMORY_SCOPE_AGENT);
        __hip_atomic_fetch_add(p + 1, a * v.y, __ATOMIC_RELAXED, __HIP_MEMORY_SCOPE_AGENT);
        __hip_atomic_fetch_add(p + 2, a * v.z, __ATOMIC_RELAXED, __HIP_MEMORY_SCOPE_AGENT);
        __hip_atomic_fetch_add(p + 3, a * v.w, __ATOMIC_RELAXED, __HIP_MEMORY_SCOPE_AGENT);
    }
}

// h = relu(out1 + bias1); yl = h@W2l + b2l ; yr = h@W2r + b2r (one wave/node)
__global__ __launch_bounds__(256) void gat_relu_l2xform(
    const float4* __restrict__ out4, const float4* __restrict__ bias4,
    const float4* __restrict__ W2l4, const float* __restrict__ b2l,
    const float4* __restrict__ W2r4, const float* __restrict__ b2r,
    float* __restrict__ yl, float* __restrict__ yr, int N)
{
    const int n = blockIdx.x * 8 + (threadIdx.x >> 5);
    if (n >= N) return;
    const int lane = threadIdx.x & 31;
    const float4* __restrict__ row = out4 + (size_t)n * Q1;
    float accl = 0.0f, accr = 0.0f;
    for (int i = lane; i < Q1; i += 32) {
        float4 v = row[i], bb = bias4[i], wl = W2l4[i], wr = W2r4[i];
        float h0 = fmaxf(v.x + bb.x, 0.f);
        float h1 = fmaxf(v.y + bb.y, 0.f);
        float h2 = fmaxf(v.z + bb.z, 0.f);
        float h3 = fmaxf(v.w + bb.w, 0.f);
        accl += h0 * wl.x + h1 * wl.y + h2 * wl.z + h3 * wl.w;
        accr += h0 * wr.x + h1 * wr.y + h2 * wr.z + h3 * wr.w;
    }
    accl = wave_sum(accl);
    accr = wave_sum(accr);
    if (lane == 0) { yl[n] = accl + b2l[0]; yr[n] = accr + b2r[0]; }
}

// Layer-2 (H=1, C=1): scalar per-edge passes
__global__ void gat_edge_logit2(const float* __restrict__ yl, const float* __restrict__ yr,
                                const float* __restrict__ att2, const int* __restrict__ ei,
                                float* __restrict__ e2, int* __restrict__ m2, int E, int ET)
{
    int e = blockIdx.x * blockDim.x + threadIdx.x;
    if (e >= ET) return;
    int s, d; edge_ends(ei, e, E, s, d);
    float lg = att2[0] * lrelu(yl[s] + yr[d]);
    e2[e] = lg;
    atomicMax(&m2[d], fmap_ord(lg));
}

__global__ void gat_edge_exp2(float* __restrict__ e2, const int* __restrict__ m2,
                              float* __restrict__ den2, const int* __restrict__ ei,
                              int E, int ET)
{
    int e = blockIdx.x * blockDim.x + threadIdx.x;
    if (e >= ET) return;
    int d = (e < E) ? ei[E + e] : e - E;
    float ex = __expf(e2[e] - funmap_ord(m2[d]));
    e2[e] = ex;
    __hip_atomic_fetch_add(&den2[d], ex, __ATOMIC_RELAXED, __HIP_MEMORY_SCOPE_AGENT);
}

__global__ void gat_edge_scatter2(const float* __restrict__ e2, const float* __restrict__ den2,
                                  const float* __restrict__ yl, const int* __restrict__ ei,
                                  float* __restrict__ acc2, int E, int ET)
{
    int e = blockIdx.x * blockDim.x + threadIdx.x;
    if (e >= ET) return;
    int s, d; edge_ends(ei, e, E, s, d);
    float alpha = e2[e] / (den2[d] + 1e-16f);
    __hip_atomic_fetch_add(&acc2[d], alpha * yl[s],
                           __ATOMIC_RELAXED, __HIP_MEMORY_SCOPE_AGENT);
}

__global__ void gat_finalize(const float* __restrict__ acc2, const float* __restrict__ bias2,
                             float* __restrict__ out, int N)
{
    int n = blockIdx.x * blockDim.x + threadIdx.x;
    if (n < N) out[n] = acc2[n] + bias2[0];
}

// ---------------------------------------------------------------------------
extern "C" void kernel_launch(void* const* d_in, const int* in_sizes, int n_in,
                              void* d_out, int out_size, void* d_ws, size_t ws_size,
                              hipStream_t stream)
{
    (void)n_in; (void)out_size; (void)ws_size;
    const float* x     = (const float*)d_in[0];
    const int*   ei    = (const int*)  d_in[1];
    const float* W1l   = (const float*)d_in[2];
    const float* b1l   = (const float*)d_in[3];
    const float* W1r   = (const float*)d_in[4];
    const float* b1r   = (const float*)d_in[5];
    const float* att1  = (const float*)d_in[6];
    const float* bias1 = (const float*)d_in[7];
    const float* W2l   = (const float*)d_in[8];
    const float* b2l   = (const float*)d_in[9];
    const float* W2r   = (const float*)d_in[10];
    const float* b2r   = (const float*)d_in[11];
    const float* att2  = (const float*)d_in[12];
    const float* bias2 = (const float*)d_in[13];
    float* out = (float*)d_out;

    const int N  = in_sizes[0] / FIN;   // 20000
    const int E  = in_sizes[1] / 2;     // 200000
    const int ET = E + N;               // + self loops

    // ---- carve workspace ----
    char* ws = (char*)d_ws;
    size_t off = 0;
    auto carve = [&](size_t bytes) -> char* {
        char* p = ws + off;
        off = (off + bytes + 255) & ~(size_t)255;
        return p;
    };
    float* XL   = (float*)carve((size_t)N * D1 * 4);
    float* XR   = (float*)carve((size_t)N * D1 * 4);
    float* OUT1 = (float*)carve((size_t)N * D1 * 4);
    float* E1   = (float*)carve((size_t)ET * H1 * 4);
    int*   M1   = (int*)  carve((size_t)N * H1 * 4);
    float* DEN1 = (float*)carve((size_t)N * H1 * 4);
    float* YL   = (float*)carve((size_t)N * 4);
    float* YR   = (float*)carve((size_t)N * 4);
    float* E2   = (float*)carve((size_t)ET * 4);
    int*   M2   = (int*)  carve((size_t)N * 4);
    float* DEN2 = (float*)carve((size_t)N * 4);
    float* ACC2 = (float*)carve((size_t)N * 4);

    // ---- init accumulators (fresh every call) ----
    {
        long long tot4 = (long long)N * Q1;
        int blocks = (int)((tot4 + 255) / 256);
        gat_init<<<blocks, 256, 0, stream>>>((float4*)OUT1, DEN1, M1, DEN2, M2, ACC2, N);
    }
    // ---- layer-1 transforms (WMMA fp32 16x16x4) ----
    {
        dim3 grid((N + 15) / 16, 30);
        gat_gemm_xlr<<<grid, 128, 0, stream>>>(x, W1l, b1l, W1r, b1r, XL, XR, N);
    }
    // ---- layer-1 attention ----
    gat_edge_logits1<<<(ET + 7) / 8, 256, 0, stream>>>(
        (const float4*)XL, (const float4*)XR, (const float4*)att1, ei, E1, M1, E, ET);
    {
        long long tot = (long long)ET * H1;
        gat_edge_exp1<<<(int)((tot + 255) / 256), 256, 0, stream>>>(E1, M1, DEN1, ei, E, ET);
    }
    gat_edge_scatter1<<<(ET + 7) / 8, 256, 0, stream>>>(
        (const float4*)XL, E1, DEN1, ei, OUT1, E, ET);
    // ---- relu + layer-2 transforms ----
    gat_relu_l2xform<<<(N + 7) / 8, 256, 0, stream>>>(
        (const float4*)OUT1, (const float4*)bias1, (const float4*)W2l, b2l,
        (const float4*)W2r, b2r, YL, YR, N);
    // ---- layer-2 attention (scalar) ----
    gat_edge_logit2<<<(ET + 255) / 256, 256, 0, stream>>>(YL, YR, att2, ei, E2, M2, E, ET);
    gat_edge_exp2<<<(ET + 255) / 256, 256, 0, stream>>>(E2, M2, DEN2, ei, E, ET);
    gat_edge_scatter2<<<(ET + 255) / 256, 256, 0, stream>>>(E2, DEN2, YL, ei, ACC2, E, ET);
    gat_finalize<<<(N + 255) / 256, 256, 0, stream>>>(ACC2, bias2, out, N);
}